// MonotonicAttentionLayer_45792941310483
// MI455X (gfx1250) — compile-verified
//
#include <hip/hip_runtime.h>
#include <hip/hip_bf16.h>
#include <math.h>

typedef __attribute__((ext_vector_type(2))) float v2f;
typedef __attribute__((ext_vector_type(8))) float v8f;

// ---------------------------------------------------------------------------
// GEMM (NT): D[m,n] = post( sum_k A[m,k] * W[n,k] )
//   A is [32 x K] row-major, supplied as two halves with row stride 1024:
//     k <  split -> A1[m*1024 + k]
//     k >= split -> A2[m*1024 + (k-split)]
//   W is [N x K] row-major (so this computes A @ W^T, matching input @ W_in.T).
// One wave32 computes one 16x16 D tile with V_WMMA_F32_16X16X4_F32.
// ISA layouts (CDNA5 ISA):
//   A 16x4 : lane l(0-15)/half h : a[v] = A[m = l][k0 + 2h + v], v=0..1
//   B 4x16 :                      b[v] = B[k0 + 2h + v][n = l] = W[n0+l][k0+2h+v]
//   C 16x16: lane l, half h      : c[v] = D[m = v + 8h][n = l], v=0..7
// grid = (N/16, 32/16), block = 32 threads (1 wave).
// ---------------------------------------------------------------------------
template <bool APPLY_TANH>
__global__ void monoattn_wmma_gemm_nt(const float* __restrict__ A1,
                                      const float* __restrict__ A2,
                                      int split, int K,
                                      const float* __restrict__ W,
                                      float* __restrict__ D, int ldD)
{
    const int lane  = threadIdx.x & 31;
    const int half  = lane >> 4;
    const int l     = lane & 15;
    const int nbase = blockIdx.x * 16;
    const int mbase = blockIdx.y * 16;

    v8f c = {};
    for (int k0 = 0; k0 < K; k0 += 4) {
        const int kk = k0 + 2 * half;            // K index for this lane-half
        const float* Arow;
        int ka;
        if (k0 < split) { Arow = A1; ka = kk; }
        else            { Arow = A2; ka = kk - split; }

        v2f a, b;
        a.x = Arow[(size_t)(mbase + l) * 1024 + ka];
        a.y = Arow[(size_t)(mbase + l) * 1024 + ka + 1];
        b.x = W[(size_t)(nbase + l) * K + kk];
        b.y = W[(size_t)(nbase + l) * K + kk + 1];

        // (neg_a, A, neg_b, B, c_mod, C, reuse_a, reuse_b)
        c = __builtin_amdgcn_wmma_f32_16x16x4_f32(false, a, false, b,
                                                  (short)0, c, false, false);
    }

#pragma unroll
    for (int v = 0; v < 8; ++v) {
        float r = c[v];
        if (APPLY_TANH) r = tanhf(r);
        D[(size_t)(mbase + v + 8 * half) * ldD + nbase + l] = r;
    }
}

// ---------------------------------------------------------------------------
// energy[s,b] = dot(source_hids[s,b,:], x[b,:]) , masked -> -1e9
// One wave32 per (s,b); lanes stride over d (coalesced); shuffle reduction.
// block = 256 (8 waves), grid = S*32/8 = 8192.
// ---------------------------------------------------------------------------
__global__ void monoattn_energy(const float* __restrict__ src,   // [S,32,1024]
                                const float* __restrict__ x,     // [32,1024]
                                const unsigned char* __restrict__ mask, // [S,32]
                                float* __restrict__ energy,      // [S,32]
                                int S)
{
    const int gw   = blockIdx.x * (blockDim.x >> 5) + (threadIdx.x >> 5);
    const int lane = threadIdx.x & 31;
    const int s = gw >> 5;
    const int b = gw & 31;
    if (s >= S) return;

    const float* sp = src + ((size_t)s * 32 + b) * 1024;
    const float* xp = x + (size_t)b * 1024;

    float sum = 0.f;
    for (int d = lane; d < 1024; d += 32)
        sum += sp[d] * xp[d];

    for (int off = 16; off > 0; off >>= 1)          // wave32 reduction
        sum += __shfl_xor(sum, off, 32);

    if (lane == 0)
        energy[s * 32 + b] = mask[s * 32 + b] ? -1e9f : sum;
}

// ---------------------------------------------------------------------------
// Sequential monotonic-attention scan along s, one lane per batch column.
// Coalesced: lane b reads energy[s*32 + b]. Writes alpha into both beta and
// alpha output regions (hard monotonic => beta == alpha).
// ---------------------------------------------------------------------------
__global__ void monoattn_scan(const float* __restrict__ energy, // [S,32]
                              float* __restrict__ beta_out,     // [S,32]
                              float* __restrict__ alpha_out,    // [S,32]
                              int S)
{
    const int b = threadIdx.x;   // 0..31
    float cp = 1.0f;             // exclusive cumprod of (1-p)
    float csum = 0.0f;           // cumsum(prev_attn / clip(cp, eps, 1))
    for (int s = 0; s < S; ++s) {
        const float e = energy[s * 32 + b];
        const float p = 1.0f / (1.0f + __expf(-e));
        const float denom = fminf(fmaxf(cp, 1e-8f), 1.0f);
        csum += ((s == 0) ? 1.0f : 0.0f) / denom;
        const float alpha = p * cp * csum;
        beta_out[s * 32 + b]  = alpha;
        alpha_out[s * 32 + b] = alpha;
        cp *= (1.0f - p);
    }
}

// ---------------------------------------------------------------------------
// Weighted context, chunked over s for parallelism:
//   part[c, b, d] = sum_{s in chunk c} source_hids[s,b,d] * beta[s,b]
// block = 256 threads covering 256 consecutive d within a single b (coalesced);
// beta chunk staged through LDS. grid = (32*1024/256, CH).
// ---------------------------------------------------------------------------
__global__ void monoattn_wctx_partial(const float* __restrict__ src,  // [S,32,1024]
                                      const float* __restrict__ beta, // [S,32]
                                      float* __restrict__ part,       // [CH,32*1024]
                                      int chunkLen)
{
    __shared__ float sbeta[128];
    const int idx = blockIdx.x * blockDim.x + threadIdx.x; // 0..32767
    const int b  = idx >> 10;           // uniform within block
    const int dd = idx & 1023;
    const int s0 = blockIdx.y * chunkLen;

    for (int t = threadIdx.x; t < chunkLen; t += blockDim.x)
        sbeta[t] = beta[(s0 + t) * 32 + b];
    __syncthreads();

    float sum = 0.f;
    for (int t = 0; t < chunkLen; ++t)
        sum += src[((size_t)(s0 + t) * 32 + b) * 1024 + dd] * sbeta[t];

    part[(size_t)blockIdx.y * 32768 + idx] = sum;
}

__global__ void monoattn_wctx_reduce(const float* __restrict__ part,
                                     float* __restrict__ wc, int CH)
{
    const int idx = blockIdx.x * blockDim.x + threadIdx.x;
    float sum = 0.f;
    for (int c = 0; c < CH; ++c)
        sum += part[(size_t)c * 32768 + idx];
    wc[idx] = sum;
}

// ---------------------------------------------------------------------------
// Launch. Input order: input[32,1024], source_hids[2048,32,1024],
// encoder_padding_mask[2048,32] (bool), W_in[1024,1024], W_out[1024,2048].
// d_out = output[32,1024] ++ beta[2048,32] ++ alpha[2048,32] (fp32).
// ---------------------------------------------------------------------------
extern "C" void kernel_launch(void* const* d_in, const int* in_sizes, int n_in,
                              void* d_out, int out_size, void* d_ws, size_t ws_size,
                              hipStream_t stream)
{
    const float*         input = (const float*)d_in[0];
    const float*         src   = (const float*)d_in[1];
    const unsigned char* mask  = (const unsigned char*)d_in[2];
    const float*         W_in  = (const float*)d_in[3];
    const float*         W_out = (const float*)d_in[4];

    float* out_output = (float*)d_out;               // [32,1024]   = 32768
    float* out_beta   = out_output + 32 * 1024;      // [2048,32]   = 65536
    float* out_alpha  = out_beta + 2048 * 32;        // [2048,32]   = 65536

    float* ws     = (float*)d_ws;
    float* x      = ws;                // 32768 floats
    float* energy = ws + 32768;        // 65536 floats
    float* wc     = ws + 98304;        // 32768 floats
    float* part   = ws + 131072;       // 16*32768 floats  (total ~2.6 MB)

    // 1) x = input @ W_in^T   (WMMA f32 16x16x4)
    monoattn_wmma_gemm_nt<false><<<dim3(64, 2), 32, 0, stream>>>(
        input, input, 1024, 1024, W_in, x, 1024);

    // 2) energy = einsum(sbd,bd->sb) + mask  (first source_hids pass, 256 MB)
    monoattn_energy<<<8192, 256, 0, stream>>>(src, x, mask, energy, 2048);

    // 3) monotonic scan -> beta/alpha
    monoattn_scan<<<1, 32, 0, stream>>>(energy, out_beta, out_alpha, 2048);

    // 4) weighted_context = einsum(sbd,sb->bd)  (second source_hids pass)
    monoattn_wctx_partial<<<dim3(128, 16), 256, 0, stream>>>(src, out_beta, part, 128);
    monoattn_wctx_reduce<<<128, 256, 0, stream>>>(part, wc, 16);

    // 5) output = tanh(concat([wc, input], 1) @ W_out^T)  (WMMA f32 16x16x4)
    monoattn_wmma_gemm_nt<true><<<dim3(64, 2), 32, 0, stream>>>(
        wc, input, 1024, 2048, W_out, out_output, 1024);
}